// GraphSAGE_CustomScatter_12524124635374
// MI455X (gfx1250) — compile-verified
//
#include <hip/hip_runtime.h>
#include <hip/hip_bf16.h>

typedef __attribute__((ext_vector_type(2))) float v2f;
typedef __attribute__((ext_vector_type(8))) float v8f;

#define IN_C   128
#define OUT_C  128
#define CAT_C  256   // 2*OUT_C

// ---------------------------------------------------------------------------
// Kernel 0: zero agg [N*128] and count [N] in workspace
// ---------------------------------------------------------------------------
__global__ void zero_ws_kernel(float* p, long long n) {
    long long i = (long long)blockIdx.x * blockDim.x + threadIdx.x;
    long long stride = (long long)gridDim.x * blockDim.x;
    for (; i < n; i += stride) p[i] = 0.0f;
}

// ---------------------------------------------------------------------------
// Kernel 1: h = x @ W1^T + b1   (fp32 WMMA 16x16x4)
// One wave computes a 32-row stripe (2 row-tiles) across all 128 output
// columns (8 col-tiles).  B fragments are loaded once per (tile, k-step) and
// reused by both row-tiles: 8 B-loads + 2 A-loads feed 16 WMMAs per k-step.
// A-frag (16x4 f32, ISA 7.12.2): lanes 0-15 row M=lane, v0=K0,v1=K1;
//                                lanes 16-31 row M=lane-16, v0=K2,v1=K3.
// B-frag mirrored with N in place of M. C/D: VGPR j = row j (j+8 hi half),
// N = lane&15.
// ---------------------------------------------------------------------------
__global__ __launch_bounds__(256)
void gemm_h_kernel(const float* __restrict__ x, const float* __restrict__ W1,
                   const float* __restrict__ b1, float* __restrict__ h, int N) {
    const int wave   = threadIdx.x >> 5;
    const int lane   = threadIdx.x & 31;
    const int laneLo = lane & 15;
    const int laneHi = lane >> 4;           // 0 or 1
    const int r0 = blockIdx.x * 256 + wave * 32;
    if (r0 >= N) return;                    // wave-uniform: EXEC stays all-1s

    const int rA0 = min(r0 + laneLo,      N - 1);
    const int rA1 = min(r0 + 16 + laneLo, N - 1);
    const float* xr0 = x + (size_t)rA0 * IN_C + 2 * laneHi;
    const float* xr1 = x + (size_t)rA1 * IN_C + 2 * laneHi;

    v8f acc[2][8];
#pragma unroll
    for (int t = 0; t < 8; ++t) {
        float bv = b1[t * 16 + laneLo];     // bias depends only on N-index
        v8f tmp = {bv, bv, bv, bv, bv, bv, bv, bv};
        acc[0][t] = tmp;
        acc[1][t] = tmp;
    }

    for (int k0 = 0; k0 < IN_C; k0 += 4) {
        v2f a0, a1;
        a0.x = xr0[k0]; a0.y = xr0[k0 + 1];
        a1.x = xr1[k0]; a1.y = xr1[k0 + 1];
#pragma unroll
        for (int t = 0; t < 8; ++t) {
            const int n = t * 16 + laneLo;  // B[k][n] = W1[n][k]
            v2f b;
            b.x = W1[(size_t)n * IN_C + k0 + 2 * laneHi];
            b.y = W1[(size_t)n * IN_C + k0 + 2 * laneHi + 1];
            acc[0][t] = __builtin_amdgcn_wmma_f32_16x16x4_f32(
                false, a0, false, b, (short)0, acc[0][t], false, false);
            acc[1][t] = __builtin_amdgcn_wmma_f32_16x16x4_f32(
                false, a1, false, b, (short)0, acc[1][t], false, false);
        }
    }

#pragma unroll
    for (int m = 0; m < 2; ++m) {
#pragma unroll
        for (int t = 0; t < 8; ++t) {
#pragma unroll
            for (int j = 0; j < 8; ++j) {
                const int row = r0 + m * 16 + j + 8 * laneHi;
                if (row < N)
                    h[(size_t)row * OUT_C + t * 16 + laneLo] = acc[m][t][j];
            }
        }
    }
}

// ---------------------------------------------------------------------------
// Kernel 2: edge scatter-add.  One wave per edge: lane k moves h[col][4k..4k+3]
// into agg[row][...] with hardware float atomics (L2-resident: agg = 51 MB
// fits the 192 MB L2, so this is L2 atomic bandwidth, not HBM-random).
// ---------------------------------------------------------------------------
__global__ __launch_bounds__(256)
void scatter_kernel(const float* __restrict__ h, const int* __restrict__ ei,
                    float* __restrict__ agg, float* __restrict__ cnt,
                    long long E) {
    long long tid = (long long)blockIdx.x * blockDim.x + threadIdx.x;
    long long e = tid >> 5;
    int lane = (int)(tid & 31);
    if (e >= E) return;
    const int r = ei[e];        // edge_index[0][e] (destination / segment id)
    const int c = ei[E + e];    // edge_index[1][e] (source of message)

    const float4 v = *(const float4*)(h + (size_t)c * OUT_C + lane * 4);
    float* dst = agg + (size_t)r * OUT_C + lane * 4;
    unsafeAtomicAdd(dst + 0, v.x);
    unsafeAtomicAdd(dst + 1, v.y);
    unsafeAtomicAdd(dst + 2, v.z);
    unsafeAtomicAdd(dst + 3, v.w);
    if (lane == 0) unsafeAtomicAdd(cnt + r, 1.0f);
}

// ---------------------------------------------------------------------------
// Kernel 3: decoded = relu(concat(h, agg/max(cnt,1))) @ Wd^T + bd
// Same 32-row WMMA tiling.  The concat boundary (k=128) is a multiple of the
// k-step, so the K loop is split into a relu(h) half and a relu(mean) half —
// no per-element branches between WMMAs.
// ---------------------------------------------------------------------------
__global__ __launch_bounds__(256)
void gemm_decode_kernel(const float* __restrict__ h, const float* __restrict__ agg,
                        const float* __restrict__ cnt, const float* __restrict__ Wd,
                        const float* __restrict__ bd, float* __restrict__ out, int N) {
    const int wave   = threadIdx.x >> 5;
    const int lane   = threadIdx.x & 31;
    const int laneLo = lane & 15;
    const int laneHi = lane >> 4;
    const int r0 = blockIdx.x * 256 + wave * 32;
    if (r0 >= N) return;                    // wave-uniform

    const int rA0 = min(r0 + laneLo,      N - 1);
    const int rA1 = min(r0 + 16 + laneLo, N - 1);
    const float inv0 = 1.0f / fmaxf(cnt[rA0], 1.0f);
    const float inv1 = 1.0f / fmaxf(cnt[rA1], 1.0f);
    const float* h0 = h + (size_t)rA0 * OUT_C + 2 * laneHi;
    const float* h1 = h + (size_t)rA1 * OUT_C + 2 * laneHi;
    const float* g0 = agg + (size_t)rA0 * OUT_C + 2 * laneHi;
    const float* g1 = agg + (size_t)rA1 * OUT_C + 2 * laneHi;

    v8f acc[2][8];
#pragma unroll
    for (int t = 0; t < 8; ++t) {
        float bv = bd[t * 16 + laneLo];
        v8f tmp = {bv, bv, bv, bv, bv, bv, bv, bv};
        acc[0][t] = tmp;
        acc[1][t] = tmp;
    }

    // --- K = 0..127 : A = relu(h) -------------------------------------------
    for (int k0 = 0; k0 < OUT_C; k0 += 4) {
        v2f a0, a1;
        a0.x = fmaxf(h0[k0], 0.0f); a0.y = fmaxf(h0[k0 + 1], 0.0f);
        a1.x = fmaxf(h1[k0], 0.0f); a1.y = fmaxf(h1[k0 + 1], 0.0f);
#pragma unroll
        for (int t = 0; t < 8; ++t) {
            const int n = t * 16 + laneLo;  // B[k][n] = Wd[n][k], Wd is [128,256]
            v2f b;
            b.x = Wd[(size_t)n * CAT_C + k0 + 2 * laneHi];
            b.y = Wd[(size_t)n * CAT_C + k0 + 2 * laneHi + 1];
            acc[0][t] = __builtin_amdgcn_wmma_f32_16x16x4_f32(
                false, a0, false, b, (short)0, acc[0][t], false, false);
            acc[1][t] = __builtin_amdgcn_wmma_f32_16x16x4_f32(
                false, a1, false, b, (short)0, acc[1][t], false, false);
        }
    }

    // --- K = 128..255 : A = relu(agg * 1/max(cnt,1)) ------------------------
    for (int k0 = 0; k0 < OUT_C; k0 += 4) {
        v2f a0, a1;
        a0.x = fmaxf(g0[k0] * inv0, 0.0f); a0.y = fmaxf(g0[k0 + 1] * inv0, 0.0f);
        a1.x = fmaxf(g1[k0] * inv1, 0.0f); a1.y = fmaxf(g1[k0 + 1] * inv1, 0.0f);
#pragma unroll
        for (int t = 0; t < 8; ++t) {
            const int n = t * 16 + laneLo;
            const int k = OUT_C + k0;       // second half of Wd's K dimension
            v2f b;
            b.x = Wd[(size_t)n * CAT_C + k + 2 * laneHi];
            b.y = Wd[(size_t)n * CAT_C + k + 2 * laneHi + 1];
            acc[0][t] = __builtin_amdgcn_wmma_f32_16x16x4_f32(
                false, a0, false, b, (short)0, acc[0][t], false, false);
            acc[1][t] = __builtin_amdgcn_wmma_f32_16x16x4_f32(
                false, a1, false, b, (short)0, acc[1][t], false, false);
        }
    }

#pragma unroll
    for (int m = 0; m < 2; ++m) {
#pragma unroll
        for (int t = 0; t < 8; ++t) {
#pragma unroll
            for (int j = 0; j < 8; ++j) {
                const int row = r0 + m * 16 + j + 8 * laneHi;
                if (row < N)
                    out[(size_t)row * IN_C + t * 16 + laneLo] = acc[m][t][j];
            }
        }
    }
}

// ---------------------------------------------------------------------------
extern "C" void kernel_launch(void* const* d_in, const int* in_sizes, int n_in,
                              void* d_out, int out_size, void* d_ws, size_t ws_size,
                              hipStream_t stream) {
    const float* x  = (const float*)d_in[0];
    const int*   ei = (const int*)d_in[1];
    const float* W1 = (const float*)d_in[2];
    const float* b1 = (const float*)d_in[3];
    const float* Wd = (const float*)d_in[4];
    const float* bd = (const float*)d_in[5];
    float* out = (float*)d_out;

    const int       N = in_sizes[0] / IN_C;     // 100000
    const long long E = in_sizes[1] / 2;        // 3200000

    // Workspace layout: h [N*128] | agg [N*128] | count [N]
    float* h   = (float*)d_ws;
    float* agg = h + (size_t)N * OUT_C;
    float* cnt = agg + (size_t)N * OUT_C;

    // 0) zero agg + count
    long long nz = (long long)N * OUT_C + N;
    zero_ws_kernel<<<2048, 256, 0, stream>>>(agg, nz);

    // 1) h = x @ W1^T + b1
    const int rowBlocks = (N + 255) / 256;
    gemm_h_kernel<<<rowBlocks, 256, 0, stream>>>(x, W1, b1, h, N);

    // 2) scatter-add messages + degree counts (1 wave per edge)
    long long scatterThreads = E * 32;
    int scatterBlocks = (int)((scatterThreads + 255) / 256);
    scatter_kernel<<<scatterBlocks, 256, 0, stream>>>(h, ei, agg, cnt, E);

    // 3) decoded = relu(concat(h, mean)) @ Wd^T + bd
    gemm_decode_kernel<<<rowBlocks, 256, 0, stream>>>(h, agg, cnt, Wd, bd, out, N);
}